// SpaceTimeMaxPooling_73899207295348
// MI455X (gfx1250) — compile-verified
//
#include <hip/hip_runtime.h>
#include <stdint.h>

// SpaceTimeMaxPooling for MI455X (gfx1250).
// x: [16, 128, 4000] f32   (treated as 2048 rows x 4000)
// neighborhood: [4000, 24] i32
// out: [16, 128, 1000] f32 (2048 rows x 1000), where output node j maps to
//      global node g = (j/250)*1000 + (j%250)   (slice commutes with max-pool)

#define B_TIMES_F 2048
#define N_NODES   4000
#define MAX_NBHD  24
#define J_OUT     1000
#define FPB       2        // rows (b*F+f) staged per block
#define BLOCK     256      // 8 wave32 waves

__global__ __launch_bounds__(BLOCK)
void stmp_kernel(const float* __restrict__ x,
                 const int*   __restrict__ nbhd,
                 float*       __restrict__ out) {
    __shared__ float tile[FPB * N_NODES];   // 32 KB

    const int tid  = threadIdx.x;
    const int row0 = blockIdx.x * FPB;      // flattened b*128+f row index

    // ---- Phase 1: async-stage FPB rows of x into LDS (CDNA5 ASYNCcnt path) ----
    // Generic shared pointer low 32 bits == LDS byte address (shared aperture).
    const unsigned lds0 = (unsigned)(uintptr_t)&tile[0];
    const float* grow0 = x + (size_t)row0 * N_NODES;

    #pragma unroll
    for (int rr = 0; rr < FPB; ++rr) {
        const float* grow = grow0 + rr * N_NODES;        // uniform -> SGPR pair
        const unsigned lrow = lds0 + rr * (N_NODES * 4);
        // 4000 floats = 1000 x b128 chunks per row
        for (int i = tid; i < N_NODES / 4; i += BLOCK) {
            unsigned laddr = lrow + i * 16;              // LDS byte address (VGPR)
            int      goff  = i * 16;                     // byte offset (VGPR)
            asm volatile("global_load_async_to_lds_b128 %0, %1, %2"
                         :: "v"(laddr), "v"(goff), "s"(grow)
                         : "memory");
        }
    }
    asm volatile("s_wait_asynccnt 0" ::: "memory");
    __syncthreads();

    // ---- Phase 2: gather + max over neighborhood, LDS-resident ----
    const float* t0 = &tile[0];
    const float* t1 = &tile[N_NODES];

    for (int j = tid; j < J_OUT; j += BLOCK) {
        const int g = (j / 250) * 1000 + (j % 250);      // kept-node index
        // 24 ints per node, 96 B, 16B-aligned -> 6 x int4 loads
        const int4* nb = (const int4*)(nbhd + g * MAX_NBHD);

        float m0 = -__builtin_huge_valf();
        float m1 = -__builtin_huge_valf();
        #pragma unroll
        for (int q = 0; q < MAX_NBHD / 4; ++q) {
            const int4 n = nb[q];
            // row 0 gathers (ds_load_b32 x4)
            m0 = fmaxf(m0, fmaxf(fmaxf(t0[n.x], t0[n.y]),
                                 fmaxf(t0[n.z], t0[n.w])));
            // row 1 gathers
            m1 = fmaxf(m1, fmaxf(fmaxf(t1[n.x], t1[n.y]),
                                 fmaxf(t1[n.z], t1[n.w])));
        }
        out[(size_t)(row0 + 0) * J_OUT + j] = m0;
        out[(size_t)(row0 + 1) * J_OUT + j] = m1;
    }
}

extern "C" void kernel_launch(void* const* d_in, const int* in_sizes, int n_in,
                              void* d_out, int out_size, void* d_ws, size_t ws_size,
                              hipStream_t stream) {
    const float* x    = (const float*)d_in[0];   // 16*128*4000 f32
    const int*   nbhd = (const int*)d_in[1];     // 4000*24 i32
    float*       out  = (float*)d_out;           // 16*128*1000 f32

    (void)in_sizes; (void)n_in; (void)out_size; (void)d_ws; (void)ws_size;

    dim3 grid(B_TIMES_F / FPB);   // 1024 blocks
    dim3 block(BLOCK);            // 256 threads = 8 wave32
    stmp_kernel<<<grid, block, 0, stream>>>(x, nbhd, out);
}